// GraphTransformer_41360535060600
// MI455X (gfx1250) — compile-verified
//
#include <hip/hip_runtime.h>
#include <math.h>

// ---------------------------------------------------------------------------
// GraphTransformer (TransformerConv x4) for MI455X / gfx1250, wave32.
// Dense QKVR projections use V_WMMA_F32_16X16X4_F32 (exact f32 matrix path).
// Sparse edge phase: 2 passes (alpha+max, fused exp/denom/message) built for
// L2 bandwidth / atomic throughput — whole working set (~90MB) fits in the
// 192MB L2, so HBM (23.3 TB/s) is never the wall.
// ---------------------------------------------------------------------------

typedef float v2f __attribute__((ext_vector_type(2)));
typedef float v8f __attribute__((ext_vector_type(8)));

#define HC 64      // H*C
#define NH 4       // heads
#define CC 16      // channels per head
#define QKVR_LD 256

// --- float atomic max via order-preserving integer mapping -----------------
__device__ __forceinline__ void atomicMaxF32(float* addr, float val) {
    // IEEE-754: non-negative floats order as signed ints; negative floats
    // order *reversed* as unsigned ints. Init value is -INF (0xFF800000).
    int iv = __float_as_int(val);
    if (iv >= 0) {
        atomicMax((int*)addr, iv);
    } else {
        atomicMin((unsigned int*)addr, (unsigned int)iv);
    }
}

// --- pack [Wq|Wk|Wv|Ws] -> 64x256, [bq|bk|bv|bs] -> 256 --------------------
__global__ void pack_w_kernel(const float* __restrict__ Wq, const float* __restrict__ bq,
                              const float* __restrict__ Wk, const float* __restrict__ bk,
                              const float* __restrict__ Wv, const float* __restrict__ bv,
                              const float* __restrict__ Ws, const float* __restrict__ bs,
                              float* __restrict__ Wp, float* __restrict__ bp) {
    int t = blockIdx.x * blockDim.x + threadIdx.x;
    if (t < 64 * 256) {
        int k = t / 256, col = t % 256;
        int sel = col >> 6, c = col & 63;
        const float* W = (sel == 0) ? Wq : (sel == 1) ? Wk : (sel == 2) ? Wv : Ws;
        Wp[t] = W[k * 64 + c];
    }
    if (t < 256) {
        int sel = t >> 6, c = t & 63;
        const float* b = (sel == 0) ? bq : (sel == 1) ? bk : (sel == 2) ? bv : bs;
        bp[t] = b[c];
    }
}

// --- QKVR GEMM: Y(n x 256) = X(n x 64) @ Wp(64 x 256) + bp ------------------
// One wave per 16x16 output tile. K=64 -> 16 chained V_WMMA_F32_16X16X4_F32.
// 32-bit A 16x4 layout: lane l (0..15) = row M, lane-half hi selects K pair
//   {2*hi, 2*hi+1}; B 4x16: lane l = col N, same K-pair striping.
// C/D: vgpr i, lane -> D[i + 8*hi][l].
__global__ void gemm_qkvr(const float* __restrict__ X,
                          const float* __restrict__ Wp,
                          const float* __restrict__ bp,
                          float* __restrict__ Y, int n) {
    const int lane   = threadIdx.x & 31;
    const int wave   = (blockIdx.x * blockDim.x + threadIdx.x) >> 5;
    const int tile_m = wave >> 4;     // 16 tiles along the 256-wide output
    const int tile_n = wave & 15;
    if (tile_m * 16 >= n) return;     // wave-uniform -> EXEC all-1s for WMMA

    const int l  = lane & 15;
    const int hi = lane >> 4;
    int row = tile_m * 16 + l;
    if (row >= n) row = n - 1;        // clamp (stores are guarded below)
    const float* xrow = X + (size_t)row * 64;
    const int col = tile_n * 16 + l;

    v8f acc = {0.f, 0.f, 0.f, 0.f, 0.f, 0.f, 0.f, 0.f};
#pragma unroll
    for (int kk = 0; kk < 16; ++kk) {
        const int kb = kk * 4 + hi * 2;
        v2f a = { xrow[kb], xrow[kb + 1] };                         // b64 load
        v2f b = { Wp[(size_t)kb * 256 + col],
                  Wp[(size_t)(kb + 1) * 256 + col] };
        acc = __builtin_amdgcn_wmma_f32_16x16x4_f32(
            /*neg_a=*/false, a, /*neg_b=*/false, b,
            /*c_mod=*/(short)0, acc, /*reuse_a=*/false, /*reuse_b=*/false);
    }
    const float bv = bp[col];
#pragma unroll
    for (int i = 0; i < 8; ++i) {
        const int m = tile_m * 16 + hi * 8 + i;
        if (m < n) Y[(size_t)m * QKVR_LD + col] = acc[i] + bv;
    }
}

// --- per-layer reset: mmax=-inf, denom=0, out_acc=0 -------------------------
__global__ void init_layer(float* __restrict__ mmax, float* __restrict__ denom,
                           float* __restrict__ out_acc, int n) {
    int i = blockIdx.x * blockDim.x + threadIdx.x;
    if (i < n * NH) { mmax[i] = -INFINITY; denom[i] = 0.f; }
    if (i < n * HC) out_acc[i] = 0.f;
}

// --- alpha[e,h] = <q[dst,h,:], k[src,h,:] + ew*We[h,:]> / sqrt(C) -----------
__global__ void edge_alpha(const int* __restrict__ src, const int* __restrict__ dst,
                           const float* __restrict__ ew,
                           const float* __restrict__ qkvr,
                           const float* __restrict__ We,
                           float* __restrict__ alpha, float* __restrict__ mmax, int E) {
    int t = blockIdx.x * blockDim.x + threadIdx.x;
    if (t >= E * NH) return;
    int e = t >> 2, h = t & 3;
    int s = src[e], d = dst[e];
    float w = ew[e];
    const float4* q  = (const float4*)(qkvr + (size_t)d * QKVR_LD + h * CC);       // Q block
    const float4* k  = (const float4*)(qkvr + (size_t)s * QKVR_LD + 64 + h * CC);  // K block
    const float4* wE = (const float4*)(We + h * CC);
    float acc = 0.f;
#pragma unroll
    for (int c = 0; c < 4; ++c) {
        float4 qv = q[c], kv = k[c], wv = wE[c];
        acc += qv.x * (kv.x + w * wv.x);
        acc += qv.y * (kv.y + w * wv.y);
        acc += qv.z * (kv.z + w * wv.z);
        acc += qv.w * (kv.w + w * wv.w);
    }
    acc *= 0.25f;                       // 1/sqrt(16)
    alpha[t] = acc;
    atomicMaxF32(mmax + (size_t)d * NH + h, acc);
}

// --- fused softmax + message ------------------------------------------------
// a = exp(alpha[e,h] - m[dst,h])  (recomputed per thread; TRANS op is far
// cheaper than a separate E*H sweep through L2).
// c4==0 thread accumulates denom; all threads scatter-add the message.
__global__ void edge_message(const int* __restrict__ src, const int* __restrict__ dst,
                             const float* __restrict__ ew,
                             const float* __restrict__ qkvr,
                             const float* __restrict__ We,
                             const float* __restrict__ alpha,
                             const float* __restrict__ mmax,
                             float* __restrict__ denom,
                             float* __restrict__ out_acc, int E) {
    int t = blockIdx.x * blockDim.x + threadIdx.x;
    if (t >= E * 16) return;
    int e  = t >> 4;
    int r  = t & 15;
    int h  = r >> 2;
    int c4 = r & 3;
    int s = src[e], d = dst[e];
    float a = __expf(alpha[(size_t)e * NH + h] - mmax[(size_t)d * NH + h]);
    if (c4 == 0) atomicAdd(denom + (size_t)d * NH + h, a);
    float w = ew[e];
    float4 v  = *(const float4*)(qkvr + (size_t)s * QKVR_LD + 128 + h * CC + c4 * 4); // V block
    float4 wv = *(const float4*)(We + h * CC + c4 * 4);
    float* o = out_acc + (size_t)d * HC + h * CC + c4 * 4;
    atomicAdd(o + 0, a * (v.x + w * wv.x));
    atomicAdd(o + 1, a * (v.y + w * wv.y));
    atomicAdd(o + 2, a * (v.z + w * wv.z));
    atomicAdd(o + 3, a * (v.w + w * wv.w));
}

// --- normalize, gated residual: y = beta*r + (1-beta)*out -------------------
__global__ void node_finalize(const float* __restrict__ qkvr,
                              const float* __restrict__ out_acc,
                              const float* __restrict__ denom,
                              const float* __restrict__ Wb,   // 192
                              float* __restrict__ yout, int n) {
    int i = blockIdx.x * blockDim.x + threadIdx.x;
    if (i >= n) return;
    float inv[NH];
#pragma unroll
    for (int h = 0; h < NH; ++h)
        inv[h] = 1.f / (denom[(size_t)i * NH + h] + 1e-16f);
    const float* rrow = qkvr + (size_t)i * QKVR_LD + 192;   // R block
    const float* orow = out_acc + (size_t)i * HC;
    float acc = 0.f;
#pragma unroll
    for (int j = 0; j < HC; ++j) {
        float o = orow[j] * inv[j >> 4];
        float r = rrow[j];
        acc += o * Wb[j] + r * Wb[64 + j] + (o - r) * Wb[128 + j];
    }
    float beta = 1.f / (1.f + __expf(-acc));
#pragma unroll
    for (int j = 0; j < HC; ++j) {
        float o = orow[j] * inv[j >> 4];
        float r = rrow[j];
        yout[(size_t)i * HC + j] = beta * r + (1.f - beta) * o;
    }
}

__global__ void add_out(const float* __restrict__ a, const float* __restrict__ b,
                        float* __restrict__ o, int n) {
    int i = blockIdx.x * blockDim.x + threadIdx.x;
    if (i < n) o[i] = a[i] + b[i];
}

// ---------------------------------------------------------------------------
static void run_layer(const float* xin, const float* Wp, const float* bp,
                      const float* We, const float* Wb,
                      const int* src, const int* dst, const float* ew,
                      float* qkvr, float* alpha, float* mmax, float* denom,
                      float* out_acc, float* yout, int n, int E, hipStream_t stream) {
    int waves  = ((n + 15) / 16) * 16;                 // 16 N-tiles per M-tile
    int blocks = (waves * 32 + 255) / 256;
    gemm_qkvr<<<blocks, 256, 0, stream>>>(xin, Wp, bp, qkvr, n);
    init_layer<<<(n * HC + 255) / 256, 256, 0, stream>>>(mmax, denom, out_acc, n);
    edge_alpha<<<(E * NH + 255) / 256, 256, 0, stream>>>(src, dst, ew, qkvr, We, alpha, mmax, E);
    edge_message<<<(E * 16 + 255) / 256, 256, 0, stream>>>(src, dst, ew, qkvr, We, alpha,
                                                           mmax, denom, out_acc, E);
    node_finalize<<<(n + 255) / 256, 256, 0, stream>>>(qkvr, out_acc, denom, Wb, yout, n);
}

extern "C" void kernel_launch(void* const* d_in, const int* in_sizes, int n_in,
                              void* d_out, int out_size, void* d_ws, size_t ws_size,
                              hipStream_t stream) {
    // Inputs in setup_inputs() dict order; params flattened in insertion order.
    const float* x   = (const float*)d_in[0];                 // N x 64
    const int*   ei  = (const int*)d_in[1];                   // 2 x E
    const float* ew  = (const float*)d_in[2];                 // E
    // params1: Wq bq Wk bk Wv bv We Ws bs Wb  (indices 3..12)
    // params2: same                            (indices 13..22)
    const float* p1[10]; const float* p2[10];
    for (int j = 0; j < 10; ++j) { p1[j] = (const float*)d_in[3 + j];
                                   p2[j] = (const float*)d_in[13 + j]; }
    const int n = in_sizes[0] / 64;
    const int E = in_sizes[2];
    const int* src = ei;
    const int* dst = ei + E;

    float* ws = (float*)d_ws;
    size_t off = 0;
    float* Wp1     = ws + off; off += 64 * 256;
    float* bp1     = ws + off; off += 256;
    float* Wp2     = ws + off; off += 64 * 256;
    float* bp2     = ws + off; off += 256;
    float* qkvr    = ws + off; off += (size_t)n * QKVR_LD;
    float* alpha   = ws + off; off += (size_t)E * NH;
    float* mmax    = ws + off; off += (size_t)n * NH;
    float* denom   = ws + off; off += (size_t)n * NH;
    float* out_acc = ws + off; off += (size_t)n * HC;
    float* buf_h   = ws + off; off += (size_t)n * HC;
    float* buf_h1  = ws + off; off += (size_t)n * HC;
    (void)ws_size; (void)n_in; (void)out_size;

    // Pack weight matrices for both parameter sets.
    pack_w_kernel<<<(64 * 256 + 255) / 256, 256, 0, stream>>>(
        p1[0], p1[1], p1[2], p1[3], p1[4], p1[5], p1[7], p1[8], Wp1, bp1);
    pack_w_kernel<<<(64 * 256 + 255) / 256, 256, 0, stream>>>(
        p2[0], p2[1], p2[2], p2[3], p2[4], p2[5], p2[7], p2[8], Wp2, bp2);

    const float* We1 = p1[6]; const float* Wb1 = p1[9];
    const float* We2 = p2[6]; const float* Wb2 = p2[9];

    // h  = layer1(x; params1)
    run_layer(x, Wp1, bp1, We1, Wb1, src, dst, ew,
              qkvr, alpha, mmax, denom, out_acc, buf_h, n, E, stream);
    // h1 = layer2(h; params2); then two more params2 passes in place
    run_layer(buf_h, Wp2, bp2, We2, Wb2, src, dst, ew,
              qkvr, alpha, mmax, denom, out_acc, buf_h1, n, E, stream);
    run_layer(buf_h1, Wp2, bp2, We2, Wb2, src, dst, ew,
              qkvr, alpha, mmax, denom, out_acc, buf_h1, n, E, stream);
    run_layer(buf_h1, Wp2, bp2, We2, Wb2, src, dst, ew,
              qkvr, alpha, mmax, denom, out_acc, buf_h1, n, E, stream);

    // out = h + h1
    add_out<<<(n * HC + 255) / 256, 256, 0, stream>>>(buf_h, buf_h1, (float*)d_out, n * HC);
}